// GraphSAGE_14920716386718
// MI455X (gfx1250) — compile-verified
//
#include <hip/hip_runtime.h>

#define N_NODES 100000
#define N_EDGES 3200000
#define IN_CH 32
#define HID 16
#define NTILES (N_NODES / 16)   // 6250 exactly -> no WMMA-wave predication needed

typedef float v2f __attribute__((ext_vector_type(2)));
typedef float v8f __attribute__((ext_vector_type(8)));

// ---------------------------------------------------------------- zero scratch
__global__ void gs_zero_kernel(float* __restrict__ p, int n) {
    int i = blockIdx.x * blockDim.x + threadIdx.x;
    int stride = gridDim.x * blockDim.x;
    for (; i < n; i += stride) p[i] = 0.0f;
}

// ------------------------------------------------- layer-1 scatter: deg + agg1
__global__ void gs_scatter1_kernel(const float* __restrict__ x,
                                   const long long* __restrict__ ei,
                                   float* __restrict__ deg,
                                   float* __restrict__ agg1) {
    int e = blockIdx.x * blockDim.x + threadIdx.x;
    if (e >= N_EDGES) return;
    int src = (int)ei[e];
    int dst = (int)ei[(size_t)N_EDGES + e];
    atomicAdd(&deg[dst], 1.0f);
    const float* xs = x + (size_t)src * IN_CH;
    float* ad = agg1 + (size_t)dst * IN_CH;
#pragma unroll
    for (int c = 0; c < IN_CH; ++c) atomicAdd(&ad[c], xs[c]);
}

// ------------------------------------------------------------ deg -> 1/max(deg,1)
__global__ void gs_deginv_kernel(float* __restrict__ deg) {
    int i = blockIdx.x * blockDim.x + threadIdx.x;
    if (i < N_NODES) deg[i] = 1.0f / fmaxf(deg[i], 1.0f);
}

// ------------------------------------------------------------- layer 1 (WMMA)
// h = relu( (agg1 * deg_inv) @ Wl1^T + b1 + x @ Wr1^T ), one wave32 per 16 nodes.
// A (16x4 f32): lanes 0-15 M=lane, {K=k0,k0+1}; lanes 16-31 M=lane-16, {K=k0+2,k0+3}
// B (4x16 f32): VGPR j holds row K=k0+j (lanes 0-15) / K=k0+2+j (lanes 16-31), N=lane&15
// C/D (16x16 f32): VGPR i holds M=i (lanes 0-15) / M=i+8 (lanes 16-31), N=lane&15
__global__ void __launch_bounds__(32)
gs_layer1_wmma_kernel(const float* __restrict__ x,
                      const float* __restrict__ agg1,
                      const float* __restrict__ deg_inv,
                      const float* __restrict__ Wl1,   // [HID][IN_CH] row-major
                      const float* __restrict__ Wr1,   // [HID][IN_CH] row-major
                      const float* __restrict__ b1,    // [HID]
                      float* __restrict__ h) {         // [N_NODES][HID]
    const int tile  = blockIdx.x;
    const int lane  = threadIdx.x;
    const int nlo   = lane & 15;        // N index for B/C/D, M index for A
    const int khalf = lane >> 4;        // which K-pair this lane holds

    const int row = tile * 16 + nlo;    // A-matrix row owned by this lane
    const float dinv = deg_inv[row];
    const float* __restrict__ xrow = x    + (size_t)row * IN_CH;
    const float* __restrict__ arow = agg1 + (size_t)row * IN_CH;

    v8f c = {0.f, 0.f, 0.f, 0.f, 0.f, 0.f, 0.f, 0.f};

#pragma unroll
    for (int kt = 0; kt < IN_CH / 4; ++kt) {
        const int k0 = kt * 4 + 2 * khalf;
        v2f a, b;

        // (deg-scaled aggregate) @ Wl1^T   -- row scaling folded into A
        a.x = arow[k0] * dinv;
        a.y = arow[k0 + 1] * dinv;
        b.x = Wl1[nlo * IN_CH + k0];        // B[k][n] = Wl1[n][k]
        b.y = Wl1[nlo * IN_CH + k0 + 1];
        c = __builtin_amdgcn_wmma_f32_16x16x4_f32(false, a, false, b,
                                                  (short)0, c, false, false);

        // x @ Wr1^T
        a.x = xrow[k0];
        a.y = xrow[k0 + 1];
        b.x = Wr1[nlo * IN_CH + k0];
        b.y = Wr1[nlo * IN_CH + k0 + 1];
        c = __builtin_amdgcn_wmma_f32_16x16x4_f32(false, a, false, b,
                                                  (short)0, c, false, false);
    }

    const float bias = b1[nlo];
#pragma unroll
    for (int i = 0; i < 8; ++i) {
        const int m = i + 8 * khalf;
        float v = c[i] + bias;
        v = v > 0.0f ? v : 0.0f;                      // ReLU
        h[((size_t)tile * 16 + m) * HID + nlo] = v;
    }
}

// ------------------------------------------------------ layer-2 scatter: agg2
__global__ void gs_scatter2_kernel(const float* __restrict__ h,
                                   const long long* __restrict__ ei,
                                   float* __restrict__ agg2) {
    int e = blockIdx.x * blockDim.x + threadIdx.x;
    if (e >= N_EDGES) return;
    int src = (int)ei[e];
    int dst = (int)ei[(size_t)N_EDGES + e];
    const float* hs = h + (size_t)src * HID;
    float* ad = agg2 + (size_t)dst * HID;
#pragma unroll
    for (int c = 0; c < HID; ++c) atomicAdd(&ad[c], hs[c]);
}

// -------------------------------------------- layer 2: length-16 dot per node
__global__ void gs_layer2_kernel(const float* __restrict__ h,
                                 const float* __restrict__ agg2,
                                 const float* __restrict__ deg_inv,
                                 const float* __restrict__ Wl2,  // [16]
                                 const float* __restrict__ Wr2,  // [16]
                                 const float* __restrict__ b2,   // [1]
                                 float* __restrict__ out) {      // [N_NODES]
    int i = blockIdx.x * blockDim.x + threadIdx.x;
    if (i >= N_NODES) return;
    const float dinv = deg_inv[i];
    const float* __restrict__ a = agg2 + (size_t)i * HID;
    const float* __restrict__ hr = h + (size_t)i * HID;
    float acc = b2[0];
#pragma unroll
    for (int c = 0; c < HID; ++c)
        acc = fmaf(a[c] * dinv, Wl2[c], fmaf(hr[c], Wr2[c], acc));
    out[i] = acc;
}

// ---------------------------------------------------------------------- launch
extern "C" void kernel_launch(void* const* d_in, const int* in_sizes, int n_in,
                              void* d_out, int out_size, void* d_ws, size_t ws_size,
                              hipStream_t stream) {
    const float*     x   = (const float*)d_in[0];
    const long long* ei  = (const long long*)d_in[1];  // int64 [2, N_EDGES]
    const float*     Wl1 = (const float*)d_in[2];
    const float*     Wr1 = (const float*)d_in[3];
    const float*     b1  = (const float*)d_in[4];
    const float*     Wl2 = (const float*)d_in[5];
    const float*     Wr2 = (const float*)d_in[6];
    const float*     b2  = (const float*)d_in[7];
    float* out = (float*)d_out;

    // Workspace layout (floats): deg | agg1 | agg2 | h
    // deg+agg1+agg2 are contiguous so one zero pass covers all accumulators.
    float* deg  = (float*)d_ws;                       // N_NODES
    float* agg1 = deg  + N_NODES;                     // N_NODES*IN_CH
    float* agg2 = agg1 + (size_t)N_NODES * IN_CH;     // N_NODES*HID
    float* h    = agg2 + (size_t)N_NODES * HID;       // N_NODES*HID (fully overwritten)

    const int zcount = N_NODES + N_NODES * IN_CH + N_NODES * HID;
    gs_zero_kernel<<<2048, 256, 0, stream>>>(deg, zcount);

    gs_scatter1_kernel<<<(N_EDGES + 255) / 256, 256, 0, stream>>>(x, ei, deg, agg1);
    gs_deginv_kernel<<<(N_NODES + 255) / 256, 256, 0, stream>>>(deg);

    gs_layer1_wmma_kernel<<<NTILES, 32, 0, stream>>>(x, agg1, deg, Wl1, Wr1, b1, h);

    gs_scatter2_kernel<<<(N_EDGES + 255) / 256, 256, 0, stream>>>(h, ei, agg2);
    gs_layer2_kernel<<<(N_NODES + 255) / 256, 256, 0, stream>>>(h, agg2, deg,
                                                                Wl2, Wr2, b2, out);
}